// iSTFT_61512521613980
// MI455X (gfx1250) — compile-verified
//
#include <hip/hip_runtime.h>

typedef __attribute__((ext_vector_type(2))) float v2f;
typedef __attribute__((ext_vector_type(8))) float v8f;

#define B_SZ 8
#define F_SZ 513      // rfft bins
#define T_SZ 4096     // frames
#define K_SZ 1024     // fft len
#define Q_SZ 4096     // output rows (hops)
#define R_SZ 512      // hop
#define N_ITERS 33    // ceil(513/16) f-chunks

// out[b, q*512 + r] = frames[b,q+1,r] (if q+1 < 4096) + frames[b,q,r+512]
// frames[b,t,k] = sum_c inputs[b,c,t] * ifft_k[c,k],
//   inputs[b,c,t] = spec[b, c%513, t, c/513]  (re rows then im rows)
__global__ __launch_bounds__(256)
void istft_wmma_kernel(const float* __restrict__ spec,
                       const float* __restrict__ ifft_k,
                       const float* __restrict__ pw,
                       float* __restrict__ out)
{
    // Transposed LDS tiles: fragment reads become single aligned b64 loads.
    __shared__ float A_ldsT[33][34];     // [t - q_base][k_local]   (33 t's: shifts 0 and 1)
    __shared__ float B_ldsT[2][64][34];  // [term][r_local][k_local]

    const int tid  = threadIdx.x;
    const int lane = tid & 31;
    const int wave = tid >> 5;
    const int ml   = lane & 15;   // low-lane index (M for A, N for B/C)
    const int hh   = lane >> 4;   // half-wave select
    const int wm   = (wave >> 2) * 16;  // wave tile row offset: 0 / 16
    const int wn   = (wave & 3) * 16;   // wave tile col offset: 0..48

    const int r_base = blockIdx.x * 64;
    const int q_base = blockIdx.y * 32;
    const int b      = blockIdx.z;

    const float2* spec2 = (const float2*)spec;   // (b,f,t) -> {re, im}

    v8f acc = {};

    for (int it = 0; it < N_ITERS; ++it) {
        const int f0 = it * 16;

        // ---- stage A: spec[b, f0..f0+15, q_base..q_base+32, :] (float2 coalesced) ----
        for (int e = tid; e < 16 * 33; e += 256) {
            int fl = e / 33;
            int tt = e - fl * 33;
            int f  = f0 + fl;
            int t  = q_base + tt;
            float2 v = make_float2(0.f, 0.f);
            if (f < F_SZ && t < T_SZ)                 // zero-pads f>=513 and t==4096 edge
                v = spec2[(size_t)(b * F_SZ + f) * T_SZ + t];
            A_ldsT[tt][fl]      = v.x;                // c = f        (real rows)
            A_ldsT[tt][16 + fl] = v.y;                // c = 513 + f  (imag rows)
        }
        // ---- stage B: ifft_k rows {f0.., 513+f0..}, cols [r_base,+64) and [512+r_base,+64) ----
        for (int e = tid; e < 4096; e += 256) {
            int term = e >> 11;          // 0: cols r_base (shift-1 term), 1: cols 512+r_base
            int rem  = e & 2047;
            int kl   = rem >> 6;         // 0..31
            int rl   = rem & 63;         // coalesced over consecutive threads
            int fidx = (kl < 16) ? (f0 + kl) : (f0 + kl - 16);
            float v = 0.f;
            if (fidx < F_SZ) {
                int c   = (kl < 16) ? fidx : (F_SZ + fidx);
                int col = (term ? (512 + r_base) : r_base) + rl;
                v = ifft_k[(size_t)c * K_SZ + col];
            }
            B_ldsT[term][rl][kl] = v;
        }
        __syncthreads();

        // ---- WMMA: 2 overlap-add terms x 8 k-steps of fp32 16x16x4 ----
        #pragma unroll
        for (int term = 0; term < 2; ++term) {
            const int shift = term ? 0 : 1;                       // term0 uses t=q+1
            const float* arow = &A_ldsT[wm + ml + shift][0];
            const float* brow = &B_ldsT[term][wn + ml][0];
            #pragma unroll
            for (int kk = 0; kk < 8; ++kk) {
                int k0 = kk * 4 + hh * 2;                         // lanes16-31 take K+2,K+3
                v2f a  = *(const v2f*)(arow + k0);
                v2f bm = *(const v2f*)(brow + k0);
                acc = __builtin_amdgcn_wmma_f32_16x16x4_f32(
                        false, a, false, bm, (short)0, acc, false, false);
            }
        }
        __syncthreads();
    }

    // ---- epilogue: window-overlap normalization, write once per element ----
    const int r = r_base + wn + ml;
    const float c_hi   = pw[r + 512];
    const float c_full = pw[r] + c_hi;
    float* outb = out + (size_t)b * Q_SZ * R_SZ;
    #pragma unroll
    for (int j = 0; j < 8; ++j) {
        int m = hh * 8 + j;                 // C/D layout: VGPR j -> M = j + 8*half
        int q = q_base + wm + m;
        float cof = (q < Q_SZ - 1) ? c_full : c_hi;
        cof = (cof > 1e-8f) ? cof : 1.f;
        outb[(size_t)q * R_SZ + r] = acc[j] / cof;
    }
}

extern "C" void kernel_launch(void* const* d_in, const int* in_sizes, int n_in,
                              void* d_out, int out_size, void* d_ws, size_t ws_size,
                              hipStream_t stream) {
    const float* spec   = (const float*)d_in[0];   // (8,513,4096,2) f32
    const float* ifft_k = (const float*)d_in[1];   // (1026,1024) f32
    const float* pw     = (const float*)d_in[2];   // (1024,) f32
    float* out = (float*)d_out;                    // (8, 2097152) f32

    dim3 grid(R_SZ / 64, Q_SZ / 32, B_SZ);         // 8 x 128 x 8
    dim3 block(256);
    hipLaunchKernelGGL(istft_wmma_kernel, grid, block, 0, stream,
                       spec, ifft_k, pw, out);
}